// SpectralTemporalMultivariableConv2d_74663711474426
// MI455X (gfx1250) — compile-verified
//
#include <hip/hip_runtime.h>
#include <hip/hip_bf16.h>

typedef __attribute__((ext_vector_type(2))) float v2f;
typedef __attribute__((ext_vector_type(8))) float v8f;

#define PI_F 3.14159265358979323846f

constexpr int IC = 8, OC = 8, NV = 3, TD = 256, BB = 16, HH = 256, WW = 256;
constexpr int KX = 32, KY = 16;   // 16 low + 16 high kx modes; 16 ky modes

__device__ __forceinline__ v8f wmma4(v2f a, v2f b, v8f c) {
  // V_WMMA_F32_16X16X4_F32: D = A(16x4) * B(4x16) + C(16x16), fp32, wave32
  return __builtin_amdgcn_wmma_f32_16x16x4_f32(false, a, false, b, (short)0, c,
                                               false, false);
}

// ---------------------------------------------------------------------------
// Kernel 1: partial rDFT along W.
// Y[b,i,v,h,ky] (complex) = sum_w x[b,i,v,h,w] * e^{-2pi i w ky / W}
// Rows R = B*IC*NV*H = 98304, K = 256, N = 16 complex (cos acc + sin acc).
// 8 waves/block, each wave owns a 16-row tile. Basis in LDS.
// ---------------------------------------------------------------------------
__global__ void dft_w_kernel(const float* __restrict__ x, float* __restrict__ Y) {
  __shared__ float lbc[WW * KY];  // cos(-theta) = cos
  __shared__ float lbs[WW * KY];  // sin(-theta) = -sin  (== Im part directly)
  int tid = threadIdx.x;
  for (int idx = tid; idx < WW * KY; idx += 256) {
    int w = idx >> 4, ky = idx & 15;
    int ph = (w * ky) & 255;
    float ang = -2.0f * PI_F * (float)ph / 256.0f;
    lbc[idx] = __cosf(ang);
    lbs[idx] = __sinf(ang);
  }
  __syncthreads();

  int wave = tid >> 5, lane = tid & 31;
  int hi = lane >> 4, mn = lane & 15;
  long rowBase = (long)blockIdx.x * 128 + wave * 16;
  const float* arow = x + (rowBase + mn) * WW;

  v8f accC = {}, accS = {};
#pragma unroll 4
  for (int k0 = 0; k0 < WW; k0 += 4) {
    int kk = k0 + 2 * hi;
    v2f a = {arow[kk], arow[kk + 1]};
    v2f bc = {lbc[kk * KY + mn], lbc[(kk + 1) * KY + mn]};
    v2f bs = {lbs[kk * KY + mn], lbs[(kk + 1) * KY + mn]};
    accC = wmma4(a, bc, accC);
    accS = wmma4(a, bs, accS);
  }
#pragma unroll
  for (int r = 0; r < 8; ++r) {
    long row = rowBase + r + 8 * hi;
    float2 val = {accC[r], accS[r]};
    *((float2*)(Y + (row * KY + mn) * 2)) = val;
  }
}

// ---------------------------------------------------------------------------
// Kernel 2: DFT along H on retained rows (small GEMM, VALU).
// Xk[b,i,v,kx,ky] = sum_h Y[b,i,v,h,ky] * e^{-2pi i h kxfull / H}
// One block per (b,i,v); Y tile staged in LDS.
// ---------------------------------------------------------------------------
__global__ void dft_h_kernel(const float* __restrict__ Y, float* __restrict__ Xk) {
  __shared__ float2 ytile[HH * KY];
  int biv = blockIdx.x;
  const float2* src = (const float2*)Y + (long)biv * HH * KY;
  int tid = threadIdx.x;
  for (int idx = tid; idx < HH * KY; idx += 256) ytile[idx] = src[idx];
  __syncthreads();

  int kx = tid >> 3;
  int kyb = (tid & 7) * 2;
  int kxfull = kx < 16 ? kx : kx + 224;  // -16..-1 -> 240..255
  float xr0 = 0.f, xi0 = 0.f, xr1 = 0.f, xi1 = 0.f;
  for (int h = 0; h < HH; ++h) {
    int ph = (h * kxfull) & 255;
    float ang = -2.0f * PI_F * (float)ph / 256.0f;
    float c = __cosf(ang), s = __sinf(ang);  // e^{-i t} = c + i s here
    float2 y0 = ytile[h * KY + kyb];
    float2 y1 = ytile[h * KY + kyb + 1];
    xr0 += c * y0.x - s * y0.y; xi0 += c * y0.y + s * y0.x;
    xr1 += c * y1.x - s * y1.y; xi1 += c * y1.y + s * y1.x;
  }
  float2* dst = (float2*)Xk + (long)biv * KX * KY;
  dst[kx * KY + kyb]     = {xr0, xi0};
  dst[kx * KY + kyb + 1] = {xr1, xi1};
}

// ---------------------------------------------------------------------------
// Kernel 3 (fused, bandwidth-critical): per spectral site (o,v,kx,ky),
//   tw[b, i(re/im)] = t[16x256] x W[256x16]  (one WMMA-tile GEMM, K=256)
//   OFT[b,o,v,kx,ky] = sum_i Xk[b,i,v,kx,ky] * tw[b,i]   (complex)
// Each weight element is read exactly once from HBM. One wave per site,
// 8 sites per block, t staged once in LDS per block.
// ---------------------------------------------------------------------------
__global__ void twmix_kernel(const float* __restrict__ t,
                             const float* __restrict__ w1r, const float* __restrict__ w1i,
                             const float* __restrict__ w2r, const float* __restrict__ w2i,
                             const float* __restrict__ Xk, float* __restrict__ OFT) {
  __shared__ float tl[BB * TD];      // 16 KB
  __shared__ float twl[8][BB * 16];  // per-wave D staging, 8 KB
  int tid = threadIdx.x;
  for (int idx = tid; idx < BB * TD; idx += 256) tl[idx] = t[idx];
  __syncthreads();

  int wave = tid >> 5, lane = tid & 31;
  int hi = lane >> 4, mn = lane & 15;
  int site = blockIdx.x * 8 + wave;
  int ky = site & 15;
  int kx = (site >> 4) & 31;
  int v  = (site >> 9) % 3;
  int o  = site / (16 * 32 * 3);

  const float* wr; const float* wi; int kxl;
  if (kx < 16) { wr = w1r; wi = w1i; kxl = kx; }
  else         { wr = w2r; wi = w2i; kxl = kx - 16; }

  // B column n: n<8 -> real weights of input channel i=n, n>=8 -> imag of i=n-8
  int icol = mn & 7;
  const float* wsel = (mn < 8) ? wr : wi;
  const float* wcol = wsel +
      ((((long)icol * OC + o) * NV + v) * 16 + kxl) * (long)(16 * TD) + (long)ky * TD;

  v8f acc = {};
#pragma unroll 4
  for (int k0 = 0; k0 < TD; k0 += 4) {
    int kk = k0 + 2 * hi;
    v2f a = {tl[mn * TD + kk], tl[mn * TD + kk + 1]};
    v2f b = {wcol[kk], wcol[kk + 1]};
    acc = wmma4(a, b, acc);
  }
#pragma unroll
  for (int r = 0; r < 8; ++r) twl[wave][(r + 8 * hi) * 16 + mn] = acc[r];
  __syncthreads();

  if (lane < 16) {
    int b = lane;
    const float2* xkb = (const float2*)Xk +
        (((long)b * IC) * NV + v) * (long)(KX * KY) + (long)kx * KY + ky;
    float orr = 0.f, oii = 0.f;
#pragma unroll
    for (int i = 0; i < 8; ++i) {
      float2 xv = xkb[(long)i * NV * KX * KY];
      float twr = twl[wave][b * 16 + i];
      float twi = twl[wave][b * 16 + 8 + i];
      orr += xv.x * twr - xv.y * twi;
      oii += xv.x * twi + xv.y * twr;
    }
    ((float2*)OFT)[(((long)(b * OC + o) * NV + v) * KX + kx) * KY + ky] = {orr, oii};
  }
}

// ---------------------------------------------------------------------------
// Kernel 4: inverse DFT along H (full complex ifft, only 32 nonzero kx).
// Z[b,o,v,h,ky] = (1/H) sum_kx OFT[..kx,ky] e^{+2pi i h kxfull / H}
// One block per (b,o,v); thread = h.
// ---------------------------------------------------------------------------
__global__ void idft_h_kernel(const float* __restrict__ OFT, float* __restrict__ Z) {
  __shared__ float2 xt[KX * KY];
  int bov = blockIdx.x;
  const float2* src = (const float2*)OFT + (long)bov * KX * KY;
  int tid = threadIdx.x;
  for (int idx = tid; idx < KX * KY; idx += 256) xt[idx] = src[idx];
  __syncthreads();

  int h = tid;
  float zr[KY] = {}, zi[KY] = {};
  for (int kx = 0; kx < KX; ++kx) {
    int kxfull = kx < 16 ? kx : kx + 224;
    int ph = (h * kxfull) & 255;
    float ang = 2.0f * PI_F * (float)ph / 256.0f;
    float c = __cosf(ang), s = __sinf(ang);
#pragma unroll
    for (int ky = 0; ky < KY; ++ky) {
      float2 xv = xt[kx * KY + ky];
      zr[ky] += c * xv.x - s * xv.y;
      zi[ky] += c * xv.y + s * xv.x;
    }
  }
  float2* dst = (float2*)Z + ((long)bov * HH + h) * KY;
  const float inv = 1.0f / 256.0f;
#pragma unroll
  for (int ky = 0; ky < KY; ++ky) dst[ky] = {zr[ky] * inv, zi[ky] * inv};
}

// ---------------------------------------------------------------------------
// Kernel 5: inverse rDFT along W (WMMA GEMM producing the 100 MB output).
// out[h,w] = sum_ky alpha_ky * (zr[h,ky] cos th - zi[h,ky] sin th),
// th = 2pi w ky / W, alpha_0 = 1/W else 2/W.
// A = z rows [16h x 32k] (k = 2*ky + {re,im}), B = basis [32k x 256w] in LDS.
// Block = (bov, h-tile); 8 waves x 2 w-tiles each.
// ---------------------------------------------------------------------------
__global__ void irdft_w_kernel(const float* __restrict__ Z, float* __restrict__ out) {
  __shared__ float bas[2 * KY * WW];  // 32 KB
  int tid = threadIdx.x;
  for (int idx = tid; idx < 2 * KY * WW; idx += 256) {
    int k = idx >> 8, w = idx & 255;
    int ky = k >> 1;
    float alpha = (ky == 0 ? 1.0f : 2.0f) / 256.0f;
    int ph = (w * ky) & 255;
    float ang = 2.0f * PI_F * (float)ph / 256.0f;
    bas[idx] = (k & 1) ? -alpha * __sinf(ang) : alpha * __cosf(ang);
  }
  __syncthreads();

  int bov = blockIdx.x >> 4;
  int ht  = blockIdx.x & 15;
  int wave = tid >> 5, lane = tid & 31;
  int hi = lane >> 4, mn = lane & 15;

  const float* arow = Z + (((long)bov * HH + ht * 16 + mn) * 2 * KY);
  int wb0 = (wave * 2) * 16;
  int wb1 = wb0 + 16;

  v8f acc0 = {}, acc1 = {};
#pragma unroll
  for (int k0 = 0; k0 < 2 * KY; k0 += 4) {
    int kk = k0 + 2 * hi;
    v2f a  = {arow[kk], arow[kk + 1]};
    v2f b0 = {bas[kk * WW + wb0 + mn], bas[(kk + 1) * WW + wb0 + mn]};
    v2f b1 = {bas[kk * WW + wb1 + mn], bas[(kk + 1) * WW + wb1 + mn]};
    acc0 = wmma4(a, b0, acc0);
    acc1 = wmma4(a, b1, acc1);
  }

  float* obase = out + (long)bov * HH * WW + (long)(ht * 16) * WW;
#pragma unroll
  for (int r = 0; r < 8; ++r) {
    int hrow = r + 8 * hi;
    obase[hrow * WW + wb0 + mn] = acc0[r];
    obase[hrow * WW + wb1 + mn] = acc1[r];
  }
}

// ---------------------------------------------------------------------------
extern "C" void kernel_launch(void* const* d_in, const int* in_sizes, int n_in,
                              void* d_out, int out_size, void* d_ws, size_t ws_size,
                              hipStream_t stream) {
  const float* t   = (const float*)d_in[0];
  const float* x   = (const float*)d_in[1];
  const float* w1r = (const float*)d_in[2];
  const float* w1i = (const float*)d_in[3];
  const float* w2r = (const float*)d_in[4];
  const float* w2i = (const float*)d_in[5];
  float* out = (float*)d_out;

  float* ws = (float*)d_ws;
  float* Y   = ws;                                       // [B,IC,NV,H,KY] cplx: 3,145,728 f
  float* Xk  = Y   + (size_t)BB * IC * NV * HH * KY * 2; // [B,IC,NV,KX,KY] cplx: 393,216 f
  float* OFT = Xk  + (size_t)BB * IC * NV * KX * KY * 2; // [B,OC,NV,KX,KY] cplx: 393,216 f
  float* Zb  = OFT + (size_t)BB * OC * NV * KX * KY * 2; // [B,OC,NV,H,KY] cplx: 3,145,728 f

  dft_w_kernel  <<<(BB * IC * NV * HH) / 128, 256, 0, stream>>>(x, Y);
  dft_h_kernel  <<<BB * IC * NV,              256, 0, stream>>>(Y, Xk);
  twmix_kernel  <<<(OC * NV * KX * KY) / 8,   256, 0, stream>>>(t, w1r, w1i, w2r, w2i, Xk, OFT);
  idft_h_kernel <<<BB * OC * NV,              256, 0, stream>>>(OFT, Zb);
  irdft_w_kernel<<<BB * OC * NV * 16,         256, 0, stream>>>(Zb, out);
}